// Generator_72610717106507
// MI455X (gfx1250) — compile-verified
//
#include <hip/hip_runtime.h>
#include <math.h>

#define VOCAB  100000
#define EMBED  100
#define HIDDEN 128
#define NOISE  10
#define BATCH  2048
#define GATES  (4*HIDDEN)     // 512
#define IN_K   (EMBED+NOISE)  // 110

#define MT 4                          // m-tiles (16 rows each) per wave
#define NT 25                         // n-tiles (16 cols each) per wave
#define MBLK (BATCH/(MT*16))          // 32
#define NSTRIPE ((VOCAB/16)/NT)       // 250
#define WAVES_PER_BLOCK 8

typedef __attribute__((ext_vector_type(2))) float v2f;
typedef __attribute__((ext_vector_type(8))) float v8f;

__device__ __forceinline__ float sigm(float x) { return 1.f / (1.f + __expf(-x)); }

// sign-aware float atomic max via integer atomics (works for mixed signs, init -inf)
__device__ __forceinline__ void atomicMaxF(float* a, float v) {
  if (v >= 0.f) atomicMax((int*)a, __float_as_int(v));
  else          atomicMin((unsigned int*)a, __float_as_uint(v));
}

// ---------------- kernel: init per-row softmax stats ----------------
__global__ void init_stats(float* rmax, float* rsum) {
  int t = blockIdx.x * blockDim.x + threadIdx.x;
  if (t < BATCH) { rmax[t] = -INFINITY; rsum[t] = 0.f; }
}

// ---------------- kernel: gates = [emb[x], noise] @ W_ih.T + b_ih + b_hh ----
__global__ void gates_kernel(const int* __restrict__ x, const float* __restrict__ noise,
                             const float* __restrict__ emb, const float* __restrict__ Wih,
                             const float* __restrict__ bih, const float* __restrict__ bhh,
                             float* __restrict__ gates) {
  int t = blockIdx.x * blockDim.x + threadIdx.x;  // [0, BATCH*GATES)
  int b = t >> 9;                                 // /512
  int r = t & (GATES - 1);
  const float* w = Wih + (long)r * IN_K;
  const float* e = emb + (long)x[b] * EMBED;
  float acc = bih[r] + bhh[r];
#pragma unroll 4
  for (int k = 0; k < EMBED; ++k) acc += e[k] * w[k];
  const float* nz = noise + b * NOISE;
#pragma unroll
  for (int k = 0; k < NOISE; ++k) acc += nz[k] * w[EMBED + k];
  gates[(long)b * GATES + r] = acc;
}

// ---------------- kernel: LSTM cell with h0=c0=0 -> h ----------------
__global__ void cell_kernel(const float* __restrict__ gates, float* __restrict__ h) {
  int t = blockIdx.x * blockDim.x + threadIdx.x;  // [0, BATCH*HIDDEN)
  int b = t >> 7;
  int j = t & (HIDDEN - 1);
  const float* g = gates + (long)b * GATES;
  float gi = g[j], gg = g[2 * HIDDEN + j], go = g[3 * HIDDEN + j];
  float c = sigm(gi) * tanhf(gg);       // c0 == 0 kills the f-gate term
  h[t] = sigm(go) * tanhf(c);
}

// ---------------- fused logits GEMM + softmax pass (3-pass recompute) -------
// MODE 0: per-row running max  -> atomicMaxF into rmax
// MODE 1: per-row sum of exp(l - rmax) -> atomicAdd into rsum
// MODE 2: write exp(l - rmax) / rsum to out
template <int MODE>
__global__ __launch_bounds__(256) void logits_kernel(
    const float* __restrict__ h, const float* __restrict__ fcw,
    const float* __restrict__ fcb, float* __restrict__ rmax,
    float* __restrict__ rsum, float* __restrict__ out) {
  const int lane = threadIdx.x;                            // 0..31 (wave32)
  const int wave = blockIdx.x * WAVES_PER_BLOCK + threadIdx.y;
  const int mblk = wave & (MBLK - 1);                      // consecutive waves share nstripe
  const int nstripe = wave / MBLK;
  const int l16 = lane & 15;
  const int hl = lane >> 4;                                // wave half
  const int koff = hl * 2;                                 // fp32 WMMA A/B K sub-offset
  const int m_base = mblk * (MT * 16);

  float rm[MT][8], sc[MT][8];
  if (MODE >= 1) {
#pragma unroll
    for (int mi = 0; mi < MT; ++mi)
#pragma unroll
      for (int i = 0; i < 8; ++i) {
        int row = m_base + mi * 16 + i + hl * 8;
        rm[mi][i] = rmax[row];
        if (MODE == 2) sc[mi][i] = 1.f / rsum[row];
      }
  }
  float st[MT][8];  // MODE0: running max; MODE1: running sum
#pragma unroll
  for (int mi = 0; mi < MT; ++mi)
#pragma unroll
    for (int i = 0; i < 8; ++i) st[mi][i] = (MODE == 0) ? -INFINITY : 0.f;

  const float* ap = h + (long)(m_base + l16) * HIDDEN + koff;

#pragma unroll 1
  for (int nt = 0; nt < NT; ++nt) {
    const int ntile = nstripe * NT + nt;
    const int col = ntile * 16 + l16;
    const float fb = fcb[col];
    const float* bp = fcw + (long)col * HIDDEN + koff;  // B[k][n] = fc_w[n][k]

    v8f acc[MT];
#pragma unroll
    for (int mi = 0; mi < MT; ++mi) acc[mi] = (v8f){0, 0, 0, 0, 0, 0, 0, 0};

#pragma unroll 4
    for (int k = 0; k < HIDDEN; k += 4) {
      v2f bfrag = *(const v2f*)(bp + k);
#pragma unroll
      for (int mi = 0; mi < MT; ++mi) {
        v2f afrag = *(const v2f*)(ap + (long)mi * 16 * HIDDEN + k);
        acc[mi] = __builtin_amdgcn_wmma_f32_16x16x4_f32(
            false, afrag, false, bfrag, (short)0, acc[mi], false, false);
      }
    }

#pragma unroll
    for (int mi = 0; mi < MT; ++mi) {
#pragma unroll
      for (int i = 0; i < 8; ++i) {
        float v = acc[mi][i] + fb;
        if (MODE == 0) {
          st[mi][i] = fmaxf(st[mi][i], v);
        } else if (MODE == 1) {
          st[mi][i] += __expf(v - rm[mi][i]);
        } else {
          int row = m_base + mi * 16 + i + hl * 8;
          out[(long)row * VOCAB + col] = __expf(v - rm[mi][i]) * sc[mi][i];
        }
      }
    }
  }

  if (MODE <= 1) {
#pragma unroll
    for (int mi = 0; mi < MT; ++mi)
#pragma unroll
      for (int i = 0; i < 8; ++i) {
        float v = st[mi][i];
#pragma unroll
        for (int off = 8; off >= 1; off >>= 1) {  // reduce within 16-lane halves
          float o = __shfl_xor(v, off, 32);
          v = (MODE == 0) ? fmaxf(v, o) : (v + o);
        }
        if (l16 == 0) {
          int row = m_base + mi * 16 + i + hl * 8;
          if (MODE == 0) atomicMaxF(&rmax[row], v);
          else           atomicAdd(&rsum[row], v);
        }
      }
  }
}

extern "C" void kernel_launch(void* const* d_in, const int* in_sizes, int n_in,
                              void* d_out, int out_size, void* d_ws, size_t ws_size,
                              hipStream_t stream) {
  (void)in_sizes; (void)n_in; (void)out_size; (void)ws_size;
  const int*   x     = (const int*)  d_in[0];
  const float* noise = (const float*)d_in[1];
  const float* emb   = (const float*)d_in[2];
  const float* Wih   = (const float*)d_in[3];
  // d_in[4] = W_hh : unused, h0 == 0
  const float* bih   = (const float*)d_in[5];
  const float* bhh   = (const float*)d_in[6];
  const float* fcw   = (const float*)d_in[7];
  const float* fcb   = (const float*)d_in[8];
  float* out = (float*)d_out;

  float* ws    = (float*)d_ws;
  float* gates = ws;                               // BATCH*GATES   floats (4 MB)
  float* hbuf  = gates + (long)BATCH * GATES;      // BATCH*HIDDEN  floats (1 MB)
  float* rmax  = hbuf + (long)BATCH * HIDDEN;      // BATCH floats
  float* rsum  = rmax + BATCH;                     // BATCH floats

  init_stats<<<(BATCH + 255) / 256, 256, 0, stream>>>(rmax, rsum);
  gates_kernel<<<(BATCH * GATES) / 256, 256, 0, stream>>>(x, noise, emb, Wih, bih, bhh, gates);
  cell_kernel<<<(BATCH * HIDDEN) / 256, 256, 0, stream>>>(gates, hbuf);

  dim3 gb(32, WAVES_PER_BLOCK);
  int nblocks = (MBLK * NSTRIPE) / WAVES_PER_BLOCK;  // 8000 waves / 8 = 1000 blocks
  logits_kernel<0><<<nblocks, gb, 0, stream>>>(hbuf, fcw, fcb, rmax, rsum, out);
  logits_kernel<1><<<nblocks, gb, 0, stream>>>(hbuf, fcw, fcb, rmax, rsum, out);
  logits_kernel<2><<<nblocks, gb, 0, stream>>>(hbuf, fcw, fcb, rmax, rsum, out);
}